// GeneralGraphConv_64132451664292
// MI455X (gfx1250) — compile-verified
//
#include <hip/hip_runtime.h>
#include <math.h>

// Problem constants (match reference)
#define N_NODES 50000
#define N_EDGES 800000
#define DIN     64
#define DOUT    64
#define NT      4
#define NR      5
#define NH      4
#define HD      16
#define LN_EPS  1e-5f

typedef float v2f __attribute__((ext_vector_type(2)));
typedef float v8f __attribute__((ext_vector_type(8)));

// ---- order-preserving float<->uint keys for atomic segment-max ----
__device__ __forceinline__ unsigned ordKey(float f) {
    unsigned u = __float_as_uint(f);
    return (u & 0x80000000u) ? ~u : (u | 0x80000000u);
}
__device__ __forceinline__ float ordVal(unsigned k) {
    unsigned u = (k & 0x80000000u) ? (k ^ 0x80000000u) : ~k;
    return __uint_as_float(u);
}

// ---------------------------------------------------------------------------
// init: zero segment-max keys, segment sums, aggregation buffer
// ---------------------------------------------------------------------------
__global__ void hgt_init(unsigned* __restrict__ mk, float* __restrict__ sm,
                         float* __restrict__ agg) {
    int i = blockIdx.x * blockDim.x + threadIdx.x;
    if (i < N_NODES * NH) { mk[i] = 0u; sm[i] = 0.f; }
    if (i < N_NODES * DOUT) { agg[i] = 0.f; }
}

// ---------------------------------------------------------------------------
// Typed K/Q/V projections. One wave (32 lanes) per 16-node tile, blockIdx.y
// selects which projection. select-by-type folded into WMMA via A-row masking:
//   out = sum_t  mask_t(x) @ W[t]      (rows of wrong type contribute 0)
// f32 WMMA 16x16x4, K=64 -> 16 k-steps; 4 col-tiles; 4 types.
// ---------------------------------------------------------------------------
__global__ __launch_bounds__(32)
void hgt_proj(const float* __restrict__ x, const int* __restrict__ ntype,
              const float* __restrict__ Wk, const float* __restrict__ bk,
              const float* __restrict__ Wq, const float* __restrict__ bq,
              const float* __restrict__ Wv, const float* __restrict__ bv,
              float* __restrict__ ws) {
    const int tile = blockIdx.x;          // 16 nodes per tile (N % 16 == 0)
    const int proj = blockIdx.y;          // 0=K 1=Q 2=V
    const int lane = threadIdx.x;
    const int sub  = lane >> 4;           // half-wave: K-pair select
    const int mrow = lane & 15;           // A-matrix row (M)
    const int node = tile * 16 + mrow;
    const int tsel = ntype[node];

    const float* W = (proj == 0) ? Wk : (proj == 1) ? Wq : Wv;
    const float* b = (proj == 0) ? bk : (proj == 1) ? bq : bv;
    float* out = ws + (size_t)proj * (size_t)N_NODES * DOUT;

    // Preload this lane's 32 A-values: d = 4*ks + 2*sub + j
    float xa[32];
#pragma unroll
    for (int ks = 0; ks < 16; ++ks) {
        int d = 4 * ks + 2 * sub;
        xa[2 * ks + 0] = x[node * DIN + d + 0];
        xa[2 * ks + 1] = x[node * DIN + d + 1];
    }

    for (int ct = 0; ct < 4; ++ct) {      // 16-wide column tiles of the output
        v8f c = {0.f, 0.f, 0.f, 0.f, 0.f, 0.f, 0.f, 0.f};
        for (int t = 0; t < NT; ++t) {
            const bool sel = (tsel == t);
            const float* Wt = W + t * DIN * DOUT;
#pragma unroll
            for (int ks = 0; ks < 16; ++ks) {
                int d = 4 * ks + 2 * sub;
                v2f a  = { sel ? xa[2 * ks + 0] : 0.f,
                           sel ? xa[2 * ks + 1] : 0.f };
                v2f bb = { Wt[(d + 0) * DOUT + ct * 16 + mrow],
                           Wt[(d + 1) * DOUT + ct * 16 + mrow] };
                c = __builtin_amdgcn_wmma_f32_16x16x4_f32(
                        false, a, false, bb, (short)0, c, false, false);
            }
        }
        // C layout: row M = v + 8*sub, col N = ct*16 + mrow
#pragma unroll
        for (int v = 0; v < 8; ++v) {
            int rn  = tile * 16 + v + 8 * sub;
            int col = ct * 16 + mrow;
            out[rn * DOUT + col] = c[v] + b[ntype[rn] * DOUT + col];
        }
    }
}

// ---------------------------------------------------------------------------
// Edge attention logits: one wave per (16-edge tile, head).
// k2 = k @ rel_att[r,h] via relation-masked f32 WMMA; att = rowwise(q . k2)
// via shfl_xor reduction within each 16-lane half; atomic segment-max.
// ---------------------------------------------------------------------------
__global__ __launch_bounds__(256)
void hgt_edge_att(const float* __restrict__ Kn, const float* __restrict__ Qn,
                  const long long* __restrict__ eidx, const int* __restrict__ etype,
                  const float* __restrict__ rel_att, const float* __restrict__ rel_pri,
                  float* __restrict__ att, unsigned* __restrict__ mk) {
    const int wid  = blockIdx.x * 8 + (threadIdx.x >> 5);
    const int tile = wid >> 2;            // 16-edge tile (E % 16 == 0)
    const int h    = wid & 3;
    const int lane = threadIdx.x & 31;
    const int sub  = lane >> 4;
    const int mrow = lane & 15;

    const int eM   = tile * 16 + mrow;    // edge owning A-row M
    const int rM   = etype[eM];
    const int srcM = (int)eidx[eM];

    // A fragment source: k_h of the source node (8 values/lane, K=16)
    const float* kb = Kn + srcM * DOUT + h * HD;
    float ka[8];
#pragma unroll
    for (int ks = 0; ks < 4; ++ks) {
        int d = 4 * ks + 2 * sub;
        ka[2 * ks + 0] = kb[d + 0];
        ka[2 * ks + 1] = kb[d + 1];
    }

    v8f c = {0.f, 0.f, 0.f, 0.f, 0.f, 0.f, 0.f, 0.f};
    for (int r = 0; r < NR; ++r) {
        const bool sel = (rM == r);
        const float* B = rel_att + ((r * NH + h) * HD) * HD;
#pragma unroll
        for (int ks = 0; ks < 4; ++ks) {
            int d = 4 * ks + 2 * sub;
            v2f a  = { sel ? ka[2 * ks + 0] : 0.f,
                       sel ? ka[2 * ks + 1] : 0.f };
            v2f bb = { B[(d + 0) * HD + mrow], B[(d + 1) * HD + mrow] };
            c = __builtin_amdgcn_wmma_f32_16x16x4_f32(
                    false, a, false, bb, (short)0, c, false, false);
        }
    }

    // att[e,h] = (q_h . k2) * rel_pri / sqrt(D);  C: (M=v+8*sub, N=mrow)
    const float scale = 0.25f;            // 1/sqrt(16)
#pragma unroll
    for (int v = 0; v < 8; ++v) {
        int e  = tile * 16 + v + 8 * sub;
        int tg = (int)eidx[N_EDGES + e];
        float p = c[v] * Qn[tg * DOUT + h * HD + mrow];
        p += __shfl_xor(p, 1, 32);
        p += __shfl_xor(p, 2, 32);
        p += __shfl_xor(p, 4, 32);
        p += __shfl_xor(p, 8, 32);        // sum over 16-lane half = sum over N
        if (mrow == 0) {
            float a = p * rel_pri[etype[e] * NH + h] * scale;
            att[e * NH + h] = a;
            atomicMax(&mk[tg * NH + h], ordKey(a));
        }
    }
}

// ---------------------------------------------------------------------------
// Edge aggregation: v2 = v @ rel_msg[r,h] via masked WMMA; ex = exp(att-max);
// atomically accumulate  sum[n,h] += ex  and  agg[n,:] += ex * v2
// (normalization by sum deferred to finalize -> identical to softmax-weighted).
// ---------------------------------------------------------------------------
__global__ __launch_bounds__(256)
void hgt_edge_agg(const float* __restrict__ Vn,
                  const long long* __restrict__ eidx, const int* __restrict__ etype,
                  const float* __restrict__ rel_msg, const float* __restrict__ att,
                  const unsigned* __restrict__ mk, float* __restrict__ sm,
                  float* __restrict__ agg) {
    const int wid  = blockIdx.x * 8 + (threadIdx.x >> 5);
    const int tile = wid >> 2;
    const int h    = wid & 3;
    const int lane = threadIdx.x & 31;
    const int sub  = lane >> 4;
    const int mrow = lane & 15;

    const int eM   = tile * 16 + mrow;
    const int rM   = etype[eM];
    const int srcM = (int)eidx[eM];

    const float* vb = Vn + srcM * DOUT + h * HD;
    float va[8];
#pragma unroll
    for (int ks = 0; ks < 4; ++ks) {
        int d = 4 * ks + 2 * sub;
        va[2 * ks + 0] = vb[d + 0];
        va[2 * ks + 1] = vb[d + 1];
    }

    v8f c = {0.f, 0.f, 0.f, 0.f, 0.f, 0.f, 0.f, 0.f};
    for (int r = 0; r < NR; ++r) {
        const bool sel = (rM == r);
        const float* B = rel_msg + ((r * NH + h) * HD) * HD;
#pragma unroll
        for (int ks = 0; ks < 4; ++ks) {
            int d = 4 * ks + 2 * sub;
            v2f a  = { sel ? va[2 * ks + 0] : 0.f,
                       sel ? va[2 * ks + 1] : 0.f };
            v2f bb = { B[(d + 0) * HD + mrow], B[(d + 1) * HD + mrow] };
            c = __builtin_amdgcn_wmma_f32_16x16x4_f32(
                    false, a, false, bb, (short)0, c, false, false);
        }
    }

    // scatter ex * v2 into agg (col = h*16 + f)
#pragma unroll
    for (int v = 0; v < 8; ++v) {
        int e  = tile * 16 + v + 8 * sub;
        int tg = (int)eidx[N_EDGES + e];
        float m  = ordVal(mk[tg * NH + h]);
        float ex = expf(att[e * NH + h] - m);
        atomicAdd(&agg[tg * DOUT + h * HD + mrow], ex * c[v]);
    }
    // one lane per edge accumulates the softmax denominator
    if (lane < 16) {
        int e  = tile * 16 + lane;
        int tg = (int)eidx[N_EDGES + e];
        float m = ordVal(mk[tg * NH + h]);
        atomicAdd(&sm[tg * NH + h], expf(att[e * NH + h] - m));
    }
}

// ---------------------------------------------------------------------------
// Finalize: normalize agg, gelu, typed Wa GEMM (masked WMMA), skip mix,
// per-row LayerNorm via shfl reductions, per-type scale/shift. One wave per
// 16-node tile.
// ---------------------------------------------------------------------------
__global__ __launch_bounds__(32)
void hgt_fin(const float* __restrict__ x, const int* __restrict__ ntype,
             const float* __restrict__ Wa, const float* __restrict__ ba,
             const float* __restrict__ lng, const float* __restrict__ lnb,
             const float* __restrict__ skipw, const float* __restrict__ sm,
             const float* __restrict__ agg, float* __restrict__ out) {
    const int tile = blockIdx.x;
    const int lane = threadIdx.x;
    const int sub  = lane >> 4;
    const int mrow = lane & 15;
    const int node = tile * 16 + mrow;
    const int tsel = ntype[node];

    // g = gelu(agg / den), per-lane A values (32 of the 64 K-elements)
    float ga[32];
#pragma unroll
    for (int ks = 0; ks < 16; ++ks) {
        int d = 4 * ks + 2 * sub;
#pragma unroll
        for (int j = 0; j < 2; ++j) {
            int dd = d + j;
            float den = sm[node * NH + (dd >> 4)];
            float a   = (den > 0.f) ? (agg[node * DOUT + dd] / den) : 0.f;
            ga[2 * ks + j] = 0.5f * a * (1.f + erff(a * 0.70710678f));
        }
    }

    v8f c[4];
    for (int ct = 0; ct < 4; ++ct) {
        v8f acc = {0.f, 0.f, 0.f, 0.f, 0.f, 0.f, 0.f, 0.f};
        for (int t = 0; t < NT; ++t) {
            const bool sel = (tsel == t);
            const float* Wt = Wa + t * DOUT * DOUT;
#pragma unroll
            for (int ks = 0; ks < 16; ++ks) {
                int d = 4 * ks + 2 * sub;
                v2f a  = { sel ? ga[2 * ks + 0] : 0.f,
                           sel ? ga[2 * ks + 1] : 0.f };
                v2f bb = { Wt[(d + 0) * DOUT + ct * 16 + mrow],
                           Wt[(d + 1) * DOUT + ct * 16 + mrow] };
                acc = __builtin_amdgcn_wmma_f32_16x16x4_f32(
                          false, a, false, bb, (short)0, acc, false, false);
            }
        }
        c[ct] = acc;
    }

    // Per output row: skip mix + LayerNorm (reduce over 64 cols = 4 ct x 16 lanes)
#pragma unroll
    for (int v = 0; v < 8; ++v) {
        int rn = tile * 16 + v + 8 * sub;
        int tt = ntype[rn];
        float sk = 1.f / (1.f + expf(-skipw[tt]));
        float res4[4];
        float s1 = 0.f, s2 = 0.f;
#pragma unroll
        for (int ct = 0; ct < 4; ++ct) {
            int col  = ct * 16 + mrow;
            float tr = c[ct][v] + ba[tt * DOUT + col];
            float rv = tr * sk + x[rn * DIN + col] * (1.f - sk);
            res4[ct] = rv;
            s1 += rv;
            s2 += rv * rv;
        }
        s1 += __shfl_xor(s1, 1, 32); s2 += __shfl_xor(s2, 1, 32);
        s1 += __shfl_xor(s1, 2, 32); s2 += __shfl_xor(s2, 2, 32);
        s1 += __shfl_xor(s1, 4, 32); s2 += __shfl_xor(s2, 4, 32);
        s1 += __shfl_xor(s1, 8, 32); s2 += __shfl_xor(s2, 8, 32);
        float mu  = s1 * (1.f / 64.f);
        float var = s2 * (1.f / 64.f) - mu * mu;
        float inv = 1.f / sqrtf(var + LN_EPS);
#pragma unroll
        for (int ct = 0; ct < 4; ++ct) {
            int col = ct * 16 + mrow;
            out[rn * DOUT + col] =
                (res4[ct] - mu) * inv * lng[tt * DOUT + col] + lnb[tt * DOUT + col];
        }
    }
}

// ---------------------------------------------------------------------------
extern "C" void kernel_launch(void* const* d_in, const int* in_sizes, int n_in,
                              void* d_out, int out_size, void* d_ws, size_t ws_size,
                              hipStream_t stream) {
    const float*     x     = (const float*)d_in[0];
    const int*       ntype = (const int*)d_in[1];
    const long long* eidx  = (const long long*)d_in[2];   // [2,E] int64
    const int*       etype = (const int*)d_in[3];
    const float* Wk = (const float*)d_in[4];  const float* bk = (const float*)d_in[5];
    const float* Wq = (const float*)d_in[6];  const float* bq = (const float*)d_in[7];
    const float* Wv = (const float*)d_in[8];  const float* bv = (const float*)d_in[9];
    const float* Wa = (const float*)d_in[10]; const float* ba = (const float*)d_in[11];
    const float* lng = (const float*)d_in[12]; const float* lnb = (const float*)d_in[13];
    const float* pri = (const float*)d_in[14]; const float* ratt = (const float*)d_in[15];
    const float* rmsg = (const float*)d_in[16]; const float* skw = (const float*)d_in[17];
    float* out = (float*)d_out;

    // workspace layout (floats): Kn | Qn | Vn | att | maxkey | sum | agg  (~66 MB)
    float* w = (float*)d_ws;
    const size_t NN  = (size_t)N_NODES * DOUT;
    const size_t EH  = (size_t)N_EDGES * NH;
    const size_t NHn = (size_t)N_NODES * NH;
    float*    Kn  = w;
    float*    Qn  = w + NN;
    float*    Vn  = w + 2 * NN;
    float*    att = w + 3 * NN;
    unsigned* mk  = (unsigned*)(w + 3 * NN + EH);
    float*    sm  = w + 3 * NN + EH + NHn;
    float*    agg = w + 3 * NN + EH + 2 * NHn;

    hgt_init<<<(int)((NN + 255) / 256), 256, 0, stream>>>(mk, sm, agg);
    hgt_proj<<<dim3(N_NODES / 16, 3), 32, 0, stream>>>(x, ntype, Wk, bk, Wq, bq,
                                                       Wv, bv, w);
    const int edgeBlocks = (N_EDGES / 16) * NH / 8;   // 8 waves/block
    hgt_edge_att<<<edgeBlocks, 256, 0, stream>>>(Kn, Qn, eidx, etype, ratt, pri,
                                                 att, mk);
    hgt_edge_agg<<<edgeBlocks, 256, 0, stream>>>(Vn, eidx, etype, rmsg, att, mk,
                                                 sm, agg);
    hgt_fin<<<N_NODES / 16, 32, 0, stream>>>(x, ntype, Wa, ba, lng, lnb, skw,
                                             sm, agg, out);
}